// GCN_22917945492092
// MI455X (gfx1250) — compile-verified
//
#include <hip/hip_runtime.h>
#include <hip/hip_bf16.h>

typedef __attribute__((ext_vector_type(2))) float v2f;
typedef __attribute__((ext_vector_type(8))) float v8f;

static inline int cdiv(long long a, long long b) { return (int)((a + b - 1) / b); }

// ---------------- degree / norm ----------------
__global__ void deg_init_kernel(float* deg, int n) {
    int i = blockIdx.x * blockDim.x + threadIdx.x;
    if (i < n) deg[i] = 1.0f;  // self-loop contributes 1
}

__global__ void deg_accum_kernel(const int* __restrict__ dst, int E, float* deg) {
    int e = blockIdx.x * blockDim.x + threadIdx.x;
    if (e < E) atomicAdd(&deg[dst[e]], 1.0f);
}

__global__ void deg_rsqrt_kernel(float* deg, int n) {
    int i = blockIdx.x * blockDim.x + threadIdx.x;
    if (i < n) deg[i] = rsqrtf(deg[i]);
}

// ---------------- fp32 WMMA GEMM, fused self-loop epilogue ----------------
// h_out[M,N]    = X[M,K] @ W[K,N]                       (raw, for edge gather)
// self_out[M,N] = h_out * dinv[row]^2 + bias[col]       (aggregation seed)
// One wave per 16x16 tile; V_WMMA_F32_16X16X4_F32, K fully unrolled.
// A frag: lane l -> A[l&15][2*(l>>4)+v] ; B frag: lane l -> B[2*(l>>4)+v][l&15]
// C/D: vgpr r -> row r + 8*(l>>4), col l&15
template <int K, int N>
__global__ void wmma_gemm_fused_kernel(const float* __restrict__ X, const float* __restrict__ W,
                                       const float* __restrict__ bias, const float* __restrict__ dinv,
                                       float* __restrict__ h_out, float* __restrict__ self_out,
                                       int M) {
    constexpr int NT = (N + 15) / 16;
    const int mt = M >> 4;  // M is a multiple of 16
    const int wave = (int)((blockIdx.x * blockDim.x + threadIdx.x) >> 5);
    if (wave >= mt * NT) return;  // whole-wave exit: EXEC stays all-ones for live waves
    const int mtile = wave / NT;
    const int ntile = wave - mtile * NT;
    const int lane = threadIdx.x & 31;
    const int half = lane >> 4;  // 0 or 1
    const int lm   = lane & 15;

    const int row = mtile * 16 + lm;
    const int col = ntile * 16 + lm;
    const bool colOK = (N % 16 == 0) ? true : (col < N);
    const int  colC  = (N % 16 == 0) ? col : (colOK ? col : 0);  // clamped address, no branch

    const float* xptr = X + (size_t)row * K + half * 2;
    const float* wptr = W + (size_t)(half * 2) * N + colC;

    v8f acc = {};
#pragma unroll
    for (int k = 0; k < K; k += 4) {
        v2f a = *(const v2f*)(xptr + k);          // global_load_b64, coalesced
        v2f b;
        b.x = wptr[(size_t)k * N];
        b.y = wptr[(size_t)k * N + N];
        if (N % 16 != 0) {                         // compile-time; pure v_cndmask select
            b.x = colOK ? b.x : 0.0f;
            b.y = colOK ? b.y : 0.0f;
        }
        acc = __builtin_amdgcn_wmma_f32_16x16x4_f32(false, a, false, b,
                                                    (short)0, acc, false, false);
    }

    if (colOK) {
        const float bv = bias[col];
#pragma unroll
        for (int r = 0; r < 8; ++r) {
            const int orow = mtile * 16 + half * 8 + r;
            const float h = acc[r];
            const float dv = dinv[orow];
            h_out[(size_t)orow * N + col]    = h;
            self_out[(size_t)orow * N + col] = h * dv * dv + bv;
        }
    }
}

// ---------------- edge aggregation ----------------
// One wave per edge; lane handles PER consecutive floats. F = 32*PER.
template <int PER>
__global__ void edge_agg_wave_kernel(const int* __restrict__ src, const int* __restrict__ dst,
                                     const float* __restrict__ dinv, const float* __restrict__ h,
                                     float* __restrict__ out, int E) {
    const int wave = (int)((blockIdx.x * blockDim.x + threadIdx.x) >> 5);
    if (wave >= E) return;
    const int lane = threadIdx.x & 31;
    const int s = src[wave], d = dst[wave];
    const float norm = dinv[s] * dinv[d];
    const int F = PER * 32;
    const float* hp = h + (size_t)s * F + lane * PER;
    float* op = out + (size_t)d * F + lane * PER;
    float v[PER];
#pragma unroll
    for (int i = 0; i < PER; ++i) v[i] = hp[i];   // coalesced b64/b128 gather
#pragma unroll
    for (int i = 0; i < PER; ++i) atomicAdd(op + i, v[i] * norm);
}

// One thread per edge, F=2 (final layer)
__global__ void edge_agg_small_kernel(const int* __restrict__ src, const int* __restrict__ dst,
                                      const float* __restrict__ dinv, const float* __restrict__ h,
                                      float* __restrict__ out, int E) {
    int e = blockIdx.x * blockDim.x + threadIdx.x;
    if (e >= E) return;
    int s = src[e], d = dst[e];
    float norm = dinv[s] * dinv[d];
    atomicAdd(&out[(size_t)d * 2 + 0], h[(size_t)s * 2 + 0] * norm);
    atomicAdd(&out[(size_t)d * 2 + 1], h[(size_t)s * 2 + 1] * norm);
}

// ---------------- epilogues ----------------
__global__ void bn_relu_kernel(float* __restrict__ h, const float* __restrict__ gamma,
                               const float* __restrict__ beta, const float* __restrict__ mean,
                               const float* __restrict__ var, int total) {
    int idx = blockIdx.x * blockDim.x + threadIdx.x;
    if (idx >= total) return;
    int f = idx & 127;  // F = 128
    float v = (h[idx] - mean[f]) * rsqrtf(var[f] + 1e-5f) * gamma[f] + beta[f];
    h[idx] = fmaxf(v, 0.0f);
}

__global__ void relu_kernel(float* __restrict__ h, int total) {
    int idx = blockIdx.x * blockDim.x + threadIdx.x;
    if (idx < total) h[idx] = fmaxf(h[idx], 0.0f);
}

// ---------------- launcher ----------------
extern "C" void kernel_launch(void* const* d_in, const int* in_sizes, int n_in,
                              void* d_out, int out_size, void* d_ws, size_t ws_size,
                              hipStream_t stream) {
    const float* x  = (const float*)d_in[0];
    const int*   ei = (const int*)d_in[1];
    const float* W1 = (const float*)d_in[2];
    const float* b1 = (const float*)d_in[3];
    const float* W2 = (const float*)d_in[4];
    const float* b2 = (const float*)d_in[5];
    const float* W3 = (const float*)d_in[6];
    const float* b3 = (const float*)d_in[7];
    const float* gm = (const float*)d_in[8];
    const float* bt = (const float*)d_in[9];
    const float* mn = (const float*)d_in[10];
    const float* vr = (const float*)d_in[11];
    float* out = (float*)d_out;

    const int n = in_sizes[0] / 128;   // 50000
    const int E = in_sizes[1] / 2;     // 800000
    const int* src = ei;
    const int* dst = ei + E;

    // workspace layout (256B aligned slices), ~51.4 MB total
    auto align_up = [](size_t v) { return (v + 255) & ~(size_t)255; };
    char* ws = (char*)d_ws;
    size_t off = 0;
    float* dinv = (float*)(ws + off); off = align_up(off + (size_t)n * sizeof(float));
    float* bufA = (float*)(ws + off); off = align_up(off + (size_t)n * 128 * sizeof(float));
    float* bufB = (float*)(ws + off); off = align_up(off + (size_t)n * 128 * sizeof(float));
    (void)ws_size;

    const int T = 256;

    // degrees -> dinv (in-degree + self-loop, then rsqrt)
    deg_init_kernel<<<cdiv(n, T), T, 0, stream>>>(dinv, n);
    deg_accum_kernel<<<cdiv(E, T), T, 0, stream>>>(dst, E, dinv);
    deg_rsqrt_kernel<<<cdiv(n, T), T, 0, stream>>>(dinv, n);

    // ---- Layer 1: GCNConv(128->128) + BN + ReLU ----
    {
        long long waves = (long long)(n / 16) * (128 / 16);
        // raw h -> bufA ; self-loop seed (h*dinv^2 + b1) -> bufB
        wmma_gemm_fused_kernel<128, 128><<<cdiv(waves * 32, T), T, 0, stream>>>(
            x, W1, b1, dinv, bufA, bufB, n);
        edge_agg_wave_kernel<4><<<cdiv((long long)E * 32, T), T, 0, stream>>>(
            src, dst, dinv, bufA, bufB, E);
        bn_relu_kernel<<<cdiv((long long)n * 128, T), T, 0, stream>>>(bufB, gm, bt, mn, vr, n * 128);
    }

    // ---- Layer 2: GCNConv(128->64) + ReLU ----
    float* h2raw = bufA;                          // [n,64]
    float* h2agg = bufA + (size_t)n * 64;         // [n,64] (second half of bufA)
    {
        long long waves = (long long)(n / 16) * (64 / 16);
        wmma_gemm_fused_kernel<128, 64><<<cdiv(waves * 32, T), T, 0, stream>>>(
            bufB, W2, b2, dinv, h2raw, h2agg, n);
        edge_agg_wave_kernel<2><<<cdiv((long long)E * 32, T), T, 0, stream>>>(
            src, dst, dinv, h2raw, h2agg, E);
        relu_kernel<<<cdiv((long long)n * 64, T), T, 0, stream>>>(h2agg, n * 64);
    }

    // ---- Layer 3: GCNConv(64->2) -> d_out ----
    float* h3raw = bufB;                          // reuse (L1 activations dead now)
    {
        long long waves = (long long)(n / 16);    // N=2 -> one 16-wide tile, select-masked
        wmma_gemm_fused_kernel<64, 2><<<cdiv(waves * 32, T), T, 0, stream>>>(
            h2agg, W3, b3, dinv, h3raw, out, n);
        edge_agg_small_kernel<<<cdiv(E, T), T, 0, stream>>>(src, dst, dinv, h3raw, out, E);
    }
}